// PredCodingSpanExtraction_35416300323649
// MI455X (gfx1250) — compile-verified
//
#include <hip/hip_runtime.h>
#include <math.h>

// Problem constants (match reference)
#define BB   256   // batch
#define SS   512   // seq len
#define DD   256   // hidden
#define LLN  4     // layers
#define NITERS 10
#define BT   16    // batch rows per workgroup in refine kernel
#define LSTR 260   // LDS row stride (pad 256 -> 260 to avoid bank conflicts)
#define TILE (BT * LSTR)
#define SMEM_FLOATS (9 * TILE)
#define SMEM_BYTES  (SMEM_FLOATS * sizeof(float))

typedef float v2f __attribute__((ext_vector_type(2)));
typedef float v8f __attribute__((ext_vector_type(8)));

#if defined(__has_builtin)
#  if __has_builtin(__builtin_amdgcn_wmma_f32_16x16x4_f32)
#    define HAS_WMMA_F32 1
#  else
#    define HAS_WMMA_F32 0
#  endif
#  if __has_builtin(__builtin_amdgcn_tanhf)
#    define HAS_HW_TANH 1
#  else
#    define HAS_HW_TANH 0
#  endif
#else
#  define HAS_WMMA_F32 0
#  define HAS_HW_TANH 0
#endif

// Hardware V_TANH_F32 (CDNA5 trans op) when available; libm fallback.
__device__ __forceinline__ float tanh_fast(float x) {
#if HAS_HW_TANH
  return __builtin_amdgcn_tanhf(x);
#else
  return tanhf(x);
#endif
}

// LDS buffer accessor: computed pointer arithmetic only, never a pointer array
// (arrays of LDS-derived pointers become illegal static initializers).
__device__ __forceinline__ float* smb(float* sm, int i) { return sm + i * TILE; }

// ---------------------------------------------------------------------------
// WMMA 16x16 tile helpers (wave32). CDNA5 f32 WMMA: D(16x16) = A(16x4)*B(4x16)+C
// A frag: lane holds row m=lane%16, k = 2*(lane/16) + {0,1} per VGPR pair
// B frag: lane holds col n=lane%16, k = 2*(lane/16) + {0,1}
// C/D   : lane holds col n=lane%16, row m = vgpr + 8*(lane/16)
// cinit: per-lane initial accumulator value (bias[n] broadcast; column is
// constant per lane so D = X@W^T + bias falls out of the MMA directly).
// ---------------------------------------------------------------------------

// Y = X @ W^T  (torch Linear): B[k][n] = W[n][k]  -> read rows of W
__device__ __forceinline__ v8f tile_xWT(const float* __restrict__ W,
                                        const float* X, int nt, int lane,
                                        float cinit) {
  const int l16 = lane & 15;
  const int half = lane >> 4;
  v8f acc = {cinit, cinit, cinit, cinit, cinit, cinit, cinit, cinit};
#if HAS_WMMA_F32
  const float* wr = W + (nt * 16 + l16) * DD + 2 * half;
  const float* xr = X + l16 * LSTR + 2 * half;
#pragma unroll 4
  for (int kk = 0; kk < DD; kk += 4) {
    v2f a = {xr[kk], xr[kk + 1]};
    v2f b = {wr[kk], wr[kk + 1]};
    acc = __builtin_amdgcn_wmma_f32_16x16x4_f32(false, a, false, b,
                                                (short)0, acc, false, false);
  }
#else
#pragma unroll
  for (int j = 0; j < 8; ++j) {
    const int m = j + 8 * half, n = nt * 16 + l16;
    float s = cinit;
    for (int k = 0; k < DD; ++k) s += X[m * LSTR + k] * W[n * DD + k];
    acc[j] = s;
  }
#endif
  return acc;
}

// Y = X @ M (no transpose): B[k][n] = M[k][n] -> read columns of M
__device__ __forceinline__ v8f tile_xM(const float* __restrict__ M,
                                       const float* X, int nt, int lane,
                                       float cinit) {
  const int l16 = lane & 15;
  const int half = lane >> 4;
  v8f acc = {cinit, cinit, cinit, cinit, cinit, cinit, cinit, cinit};
#if HAS_WMMA_F32
  const float* mp = M + (nt * 16 + l16) + (2 * half) * DD;
  const float* xr = X + l16 * LSTR + 2 * half;
#pragma unroll 4
  for (int kk = 0; kk < DD; kk += 4) {
    v2f a = {xr[kk], xr[kk + 1]};
    v2f b = {mp[kk * DD], mp[kk * DD + DD]};
    acc = __builtin_amdgcn_wmma_f32_16x16x4_f32(false, a, false, b,
                                                (short)0, acc, false, false);
  }
#else
#pragma unroll
  for (int j = 0; j < 8; ++j) {
    const int m = j + 8 * half, n = nt * 16 + l16;
    float s = cinit;
    for (int k = 0; k < DD; ++k) s += X[m * LSTR + k] * M[k * DD + n];
    acc[j] = s;
  }
#endif
  return acc;
}

// Full [BT=16, DD] x [DD, DD] GEMM (+bias) across 8 waves:
// wave wv owns n-tiles {2*wv, 2*wv+1}; epilogue gets (m, n, value).
template <typename Epi>
__device__ __forceinline__ void gemm_xWT(const float* __restrict__ W,
                                         const float* __restrict__ bias,
                                         const float* X, int lane, int wv,
                                         Epi epi) {
#pragma unroll
  for (int t = 0; t < 2; ++t) {
    const int nt = wv * 2 + t;
    const int l16 = lane & 15, half = lane >> 4;
    const float bv = bias ? bias[nt * 16 + l16] : 0.f;
    v8f acc = tile_xWT(W, X, nt, lane, bv);
#pragma unroll
    for (int j = 0; j < 8; ++j) epi(j + 8 * half, nt * 16 + l16, acc[j]);
  }
}

template <typename Epi>
__device__ __forceinline__ void gemm_xM(const float* __restrict__ M,
                                        const float* X, int lane, int wv,
                                        Epi epi) {
#pragma unroll
  for (int t = 0; t < 2; ++t) {
    const int nt = wv * 2 + t;
    v8f acc = tile_xM(M, X, nt, lane, 0.f);
    const int l16 = lane & 15, half = lane >> 4;
#pragma unroll
    for (int j = 0; j < 8; ++j) epi(j + 8 * half, nt * 16 + l16, acc[j]);
  }
}

// ---------------------------------------------------------------------------
// K1: pooled[b,d] = mean_s( embedding[ids[b,s]][d] + pos_enc[s][d] )
// ---------------------------------------------------------------------------
__global__ __launch_bounds__(256) void pooled_kernel(
    const int* __restrict__ ids, const float* __restrict__ emb,
    const float* __restrict__ pos, float* __restrict__ pooled_g) {
  __shared__ int sid[SS];
  const int b = blockIdx.x;
  const int t = threadIdx.x;
  sid[t] = ids[b * SS + t];
  sid[t + 256] = ids[b * SS + 256 + t];
  __syncthreads();
  float acc = 0.f;
  for (int s = 0; s < SS; ++s) {
    acc += emb[(size_t)sid[s] * DD + t] + pos[s * DD + t];
  }
  pooled_g[b * DD + t] = acc * (1.0f / (float)SS);
}

// ---------------------------------------------------------------------------
// K2: full predictive-coding refinement + heads for a 16-row batch tile.
// LDS buffer map (each TILE = 16x260 floats):
//   0: pooled, 1..4: reps[0..3], 5: ff, 6/7: pred ping-pong, 8: tot/scratch
// ---------------------------------------------------------------------------
__global__ __launch_bounds__(256) void refine_kernel(
    const float* __restrict__ pooled_g,
    const float* __restrict__ init_w, const float* __restrict__ init_b,
    const float* __restrict__ pred_w, const float* __restrict__ pred_b,
    const float* __restrict__ upd_w, const float* __restrict__ upd_b,
    const float* __restrict__ sqw, const float* __restrict__ sqb,
    const float* __restrict__ eqw, const float* __restrict__ eqb,
    const float* __restrict__ keyw, const float* __restrict__ keyb,
    const float* __restrict__ aw1, const float* __restrict__ ab1,
    const float* __restrict__ aw2, const float* __restrict__ ab2,
    float* __restrict__ sq_g, float* __restrict__ eq_g,
    float* __restrict__ cs_g, float* __restrict__ ce_g,
    float* __restrict__ ans_g) {
  extern __shared__ float sm[];

  const int tid = threadIdx.x;
  const int lane = tid & 31;
  const int wv = tid >> 5;
  const int b0 = blockIdx.x * BT;

  float* pooled = smb(sm, 0);
  float* ff = smb(sm, 5);
  float* tot = smb(sm, 8);

  // Load pooled tile into LDS
  for (int idx = tid; idx < BT * DD; idx += 256) {
    const int r = idx >> 8, c = idx & 255;
    pooled[r * LSTR + c] = pooled_g[(b0 + r) * DD + c];
  }
  __syncthreads();

  // ---- init chain: cur = tanh(cur @ init_w[l]^T + init_b[l]) ----
  {
    const float* cur = pooled;
    for (int l = 0; l < LLN; ++l) {
      float* dst = smb(sm, 1 + l);
      const bool last = (l == LLN - 1);
      gemm_xWT(init_w + l * DD * DD, init_b + l * DD, cur, lane, wv,
               [&](int m, int n, float v) {
                 const float r = tanh_fast(v);
                 dst[m * LSTR + n] = r;
                 if (last) ff[m * LSTR + n] = r;
               });
      __syncthreads();
      cur = dst;
    }
  }

  // ---- iterative refinement (descending in-place == simultaneous update) ----
  for (int it = 0; it < NITERS; ++it) {
    for (int i = LLN - 1; i >= 0; --i) {
      float* pc = smb(sm, 6 + ((LLN - 1 - i) & 1));  // pred[i] (this step)
      float* pn = smb(sm, 6 + ((LLN - i) & 1));      // pred[i+1] (prev step)
      float* ri = smb(sm, 1 + i);                    // reps[i]
      // pred[i] = reps[i] @ pred_w[i]^T + pred_b[i]  (reps[i] still old)
      gemm_xWT(pred_w + i * DD * DD, pred_b + i * DD, ri, lane, wv,
               [&](int m, int n, float v) { pc[m * LSTR + n] = v; });
      __syncthreads();
      // total_error = (below - pred[i]) + 0.5*(reps[i] - pred[i+1])
      {
        const float* below = (i > 0) ? smb(sm, i) : pooled;  // reps[i-1]|pooled
        const bool has_above = (i < LLN - 1);
        for (int idx = tid; idx < BT * DD; idx += 256) {
          const int r = idx >> 8, c = idx & 255;
          const int o = r * LSTR + c;
          float e = below[o] - pc[o];
          if (has_above) e += 0.5f * (ri[o] - pn[o]);
          tot[o] = e;
        }
      }
      __syncthreads();
      // reps[i] += 0.1 * tanh(total_error @ upd_w[i]^T + upd_b[i])
      gemm_xWT(upd_w + i * DD * DD, upd_b + i * DD, tot, lane, wv,
               [&](int m, int n, float v) {
                 const int o = m * LSTR + n;
                 ri[o] = ri[o] + 0.1f * tanh_fast(v);
               });
      __syncthreads();
    }
  }

  // refined = reps[L-1] + ff  (in place in reps[L-1])
  float* refined = smb(sm, LLN);  // reps[LLN-1]
  for (int idx = tid; idx < BT * DD; idx += 256) {
    const int r = idx >> 8, c = idx & 255;
    refined[r * LSTR + c] += ff[r * LSTR + c];
  }
  __syncthreads();

  // ---- start head: start_q = refined @ sqw^T + sqb ----
  {
    float* q = smb(sm, 6);
    gemm_xWT(sqw, sqb, refined, lane, wv,
             [&](int m, int n, float v) { q[m * LSTR + n] = v; });
    __syncthreads();
    // c_s[b] = start_q[b] . key_b
    {
      const int r = tid >> 4, c = tid & 15;
      float p = 0.f;
#pragma unroll
      for (int u = 0; u < 16; ++u) {
        const int o = c * 16 + u;
        p += q[r * LSTR + o] * keyb[o];
      }
      tot[tid] = p;
    }
    __syncthreads();
    if (tid < BT) {
      float s = 0.f;
#pragma unroll
      for (int u = 0; u < 16; ++u) s += tot[tid * 16 + u];
      cs_g[b0 + tid] = s;
    }
    __syncthreads();
    // sq = start_q @ key_w  -> global (folded query)
    gemm_xM(keyw, q, lane, wv,
            [&](int m, int n, float v) { sq_g[(b0 + m) * DD + n] = v; });
    __syncthreads();
  }

  // ---- end head ----
  {
    float* q = smb(sm, 7);
    gemm_xWT(eqw, eqb, refined, lane, wv,
             [&](int m, int n, float v) { q[m * LSTR + n] = v; });
    __syncthreads();
    {
      const int r = tid >> 4, c = tid & 15;
      float p = 0.f;
#pragma unroll
      for (int u = 0; u < 16; ++u) {
        const int o = c * 16 + u;
        p += q[r * LSTR + o] * keyb[o];
      }
      tot[tid] = p;
    }
    __syncthreads();
    if (tid < BT) {
      float s = 0.f;
#pragma unroll
      for (int u = 0; u < 16; ++u) s += tot[tid * 16 + u];
      ce_g[b0 + tid] = s;
    }
    __syncthreads();
    gemm_xM(keyw, q, lane, wv,
            [&](int m, int n, float v) { eq_g[(b0 + m) * DD + n] = v; });
    __syncthreads();
  }

  // ---- answerable head: h = gelu(refined @ aw1^T + ab1); ans = h @ aw2^T + ab2
  for (int task = tid; task < BT * 128; task += 256) {
    const int r = task >> 7, j = task & 127;
    float s = ab1[j];
    const float* rr = refined + r * LSTR;
    const float* wr = aw1 + j * DD;
    for (int k = 0; k < DD; ++k) s += rr[k] * wr[k];
    tot[r * LSTR + j] = 0.5f * s * (1.0f + erff(s * 0.70710678118654752f));
  }
  __syncthreads();
  if (tid < BT * 2) {
    const int r = tid >> 1, k = tid & 1;
    float s = ab2[k];
    for (int j = 0; j < 128; ++j) s += tot[r * LSTR + j] * aw2[k * 128 + j];
    ans_g[(b0 + r) * 2 + k] = s;
  }
}

// ---------------------------------------------------------------------------
// K3: start/end logits via folded queries:
//   logits[b,s] = (emb[ids[b,s]] + pos[s]) . q[b] + c[b]
// ---------------------------------------------------------------------------
__global__ __launch_bounds__(256) void logits_kernel(
    const int* __restrict__ ids, const float* __restrict__ emb,
    const float* __restrict__ pos, const float* __restrict__ sq_g,
    const float* __restrict__ eq_g, const float* __restrict__ cs_g,
    const float* __restrict__ ce_g, float* __restrict__ out_start,
    float* __restrict__ out_end) {
  const int b = blockIdx.x;
  const int lane = threadIdx.x & 31;
  const int wv = threadIdx.x >> 5;
  float sqv[8], eqv[8];
  const float* sr = sq_g + b * DD + lane * 8;
  const float* er = eq_g + b * DD + lane * 8;
#pragma unroll
  for (int j = 0; j < 8; ++j) {
    sqv[j] = sr[j];
    eqv[j] = er[j];
  }
  const float cs = cs_g[b];
  const float ce = ce_g[b];
  for (int s = wv; s < SS; s += 8) {
    const int id = ids[b * SS + s];
    const float* row = emb + (size_t)id * DD + lane * 8;
    const float* pr = pos + s * DD + lane * 8;
    float as = 0.f, ae = 0.f;
#pragma unroll
    for (int j = 0; j < 8; ++j) {
      const float e = row[j] + pr[j];
      as += e * sqv[j];
      ae += e * eqv[j];
    }
#pragma unroll
    for (int off = 16; off; off >>= 1) {
      as += __shfl_xor(as, off, 32);
      ae += __shfl_xor(ae, off, 32);
    }
    if (lane == 0) {
      out_start[b * SS + s] = as + cs;
      out_end[b * SS + s] = ae + ce;
    }
  }
}

// ---------------------------------------------------------------------------
extern "C" void kernel_launch(void* const* d_in, const int* in_sizes, int n_in,
                              void* d_out, int out_size, void* d_ws,
                              size_t ws_size, hipStream_t stream) {
  const int* ids = (const int*)d_in[0];
  const float* emb = (const float*)d_in[1];
  const float* pos = (const float*)d_in[2];
  const float* init_w = (const float*)d_in[3];
  const float* init_b = (const float*)d_in[4];
  const float* pred_w = (const float*)d_in[5];
  const float* pred_b = (const float*)d_in[6];
  const float* upd_w = (const float*)d_in[7];
  const float* upd_b = (const float*)d_in[8];
  const float* sqw = (const float*)d_in[9];
  const float* sqb = (const float*)d_in[10];
  const float* eqw = (const float*)d_in[11];
  const float* eqb = (const float*)d_in[12];
  const float* keyw = (const float*)d_in[13];
  const float* keyb = (const float*)d_in[14];
  const float* aw1 = (const float*)d_in[15];
  const float* ab1 = (const float*)d_in[16];
  const float* aw2 = (const float*)d_in[17];
  const float* ab2 = (const float*)d_in[18];

  float* ws = (float*)d_ws;
  float* pooled = ws;                 // B*D
  float* sq = pooled + BB * DD;       // B*D
  float* eq = sq + BB * DD;           // B*D
  float* cs = eq + BB * DD;           // B
  float* ce = cs + BB;                // B

  float* out = (float*)d_out;
  float* out_start = out;             // B*S
  float* out_end = out + BB * SS;     // B*S
  float* out_ans = out + 2 * BB * SS; // B*2

  // refine_kernel uses ~146 KB dynamic LDS (> default cap) — raise the limit.
  (void)hipFuncSetAttribute((const void*)refine_kernel,
                            hipFuncAttributeMaxDynamicSharedMemorySize,
                            (int)SMEM_BYTES);

  pooled_kernel<<<BB, 256, 0, stream>>>(ids, emb, pos, pooled);
  refine_kernel<<<BB / BT, 256, SMEM_BYTES, stream>>>(
      pooled, init_w, init_b, pred_w, pred_b, upd_w, upd_b, sqw, sqb, eqw, eqb,
      keyw, keyb, aw1, ab1, aw2, ab2, sq, eq, cs, ce, out_ans);
  logits_kernel<<<BB, 256, 0, stream>>>(ids, emb, pos, sq, eq, cs, ce,
                                        out_start, out_end);
}